// Cheby_79053168050932
// MI455X (gfx1250) — compile-verified
//
#include <hip/hip_runtime.h>
#include <hip/hip_bf16.h>

#define N_NODES 50000
#define N_EDGES 800000
#define F_FEAT 128           // prop always operates on 128-wide features
#define F_OUT_FINAL 64

typedef __attribute__((ext_vector_type(2))) float v2f;
typedef __attribute__((ext_vector_type(8))) float v8f;

// ---------------------------------------------------------------------------
// Elementwise helpers
// ---------------------------------------------------------------------------
__global__ void fill_zero_kernel(float4* __restrict__ p, int n4) {
  int i = blockIdx.x * blockDim.x + threadIdx.x;
  if (i < n4) p[i] = make_float4(0.f, 0.f, 0.f, 0.f);
}

// dstp[i] = -srcp[i]  (strictly out-of-place)
__global__ void neg_copy_kernel(float4* __restrict__ dstp,
                                const float4* __restrict__ srcp, int n4) {
  int i = blockIdx.x * blockDim.x + threadIdx.x;
  if (i >= n4) return;
  float4 v = srcp[i];
  dstp[i] = make_float4(-v.x, -v.y, -v.z, -v.w);
}

// p[i] = -p[i]  (in-place negate, no aliasing assumptions)
__global__ void neg_inplace_kernel(float4* p, int n4) {
  int i = blockIdx.x * blockDim.x + threadIdx.x;
  if (i >= n4) return;
  float4 v = p[i];
  p[i] = make_float4(-v.x, -v.y, -v.z, -v.w);
}

// ---------------------------------------------------------------------------
// Graph normalization: deg (segment-sum at src), dis = rsqrt, norm per edge
// ---------------------------------------------------------------------------
__global__ void deg_kernel(const int* __restrict__ src, const int* __restrict__ dst,
                           const float* __restrict__ w, float* __restrict__ deg) {
  int e = blockIdx.x * blockDim.x + threadIdx.x;
  if (e >= N_EDGES) return;
  int s = src[e], d = dst[e];
  float wv = (s == d) ? 0.0f : w[e];
  atomicAdd(&deg[s], wv);
}

__global__ void dis_kernel(float* __restrict__ deg_dis) {
  int i = blockIdx.x * blockDim.x + threadIdx.x;
  if (i >= N_NODES) return;
  float d = deg_dis[i];
  deg_dis[i] = (d > 0.0f) ? rsqrtf(d) : 0.0f;
}

__global__ void norm_kernel(const int* __restrict__ src, const int* __restrict__ dst,
                            const float* __restrict__ w, const float* __restrict__ dis,
                            float* __restrict__ norm) {
  int e = blockIdx.x * blockDim.x + threadIdx.x;
  if (e >= N_EDGES) return;
  int s = src[e], d = dst[e];
  float wv = (s == d) ? 0.0f : w[e];
  norm[e] = -dis[s] * wv * dis[d];
}

// ---------------------------------------------------------------------------
// prop: vout[dst] += scale * norm[e] * vin[src]   (one wave per edge, F=128)
// Feature buffer (25.6 MB) is L2-resident on the 192 MB L2; gathers are
// float4 loads, scatter is global_atomic_add_f32.
// ---------------------------------------------------------------------------
__global__ void prop_kernel(const int* __restrict__ src, const int* __restrict__ dst,
                            const float* __restrict__ norm,
                            const float* __restrict__ vin, float* __restrict__ vout,
                            float scale) {
  int gid = blockIdx.x * blockDim.x + threadIdx.x;
  int e = gid >> 5;
  int lane = gid & 31;
  if (e >= N_EDGES) return;
  float c = scale * norm[e];
  if (c == 0.0f) return;
  int s = src[e], d = dst[e];
  const float4 v = *(const float4*)(vin + (size_t)s * F_FEAT + lane * 4);
  float* o = vout + (size_t)d * F_FEAT + lane * 4;
  atomicAdd(o + 0, c * v.x);
  atomicAdd(o + 1, c * v.y);
  atomicAdd(o + 2, c * v.z);
  atomicAdd(o + 3, c * v.w);
}

// ---------------------------------------------------------------------------
// WMMA fp32 GEMM:  out[16x16 tile] (+)= T(16 x f_in) @ W(f_in x f_out)
// One wave per output tile; chained V_WMMA_F32_16X16X4_F32 over K.
// load_c: accumulate onto existing out; epilogue: += bias, ReLU.
// ---------------------------------------------------------------------------
__global__ __launch_bounds__(256)
void cheb_gemm_wmma(const float* __restrict__ T, const float* __restrict__ W,
                    const float* __restrict__ bias, float* __restrict__ out,
                    int f_in, int f_out, int load_c, int epilogue) {
  const int nT = f_out >> 4;                          // column tiles
  const int totalTiles = (N_NODES / 16) * nT;         // 50000 % 16 == 0
  const int wave = blockIdx.x * (blockDim.x >> 5) + (threadIdx.x >> 5);
  if (wave >= totalTiles) return;
  const int mTile = wave / nT;
  const int nTile = wave - mTile * nT;
  const int lane = threadIdx.x & 31;
  const int half = lane >> 4;       // 0: lanes 0-15, 1: lanes 16-31
  const int l16  = lane & 15;
  const int row0 = mTile * 16;
  const int col  = nTile * 16 + l16;

  v8f c = {};
  if (load_c) {
#pragma unroll
    for (int r = 0; r < 8; ++r)
      c[r] = out[(size_t)(row0 + r + half * 8) * f_out + col];
  }

  // A fragment (16x4 f32): lane holds A[l16][k + 2*half + {0,1}] -> aligned 8B load
  const float* arow = T + (size_t)(row0 + l16) * f_in + half * 2;
  // B fragment (4x16 f32): lane holds B[k + 2*half + {0,1}][col]
  const float* bcol = W + (size_t)(half * 2) * f_out + col;

#pragma unroll 8
  for (int k = 0; k < f_in; k += 4) {
    v2f a = *(const v2f*)(arow + k);
    v2f b;
    b.x = bcol[(size_t)k * f_out];
    b.y = bcol[(size_t)k * f_out + f_out];
    c = __builtin_amdgcn_wmma_f32_16x16x4_f32(false, a, false, b,
                                              (short)0, c, false, false);
  }

  if (epilogue) {
    const float bv = bias[col];
#pragma unroll
    for (int r = 0; r < 8; ++r) c[r] = fmaxf(c[r] + bv, 0.0f);
  }
#pragma unroll
  for (int r = 0; r < 8; ++r)
    out[(size_t)(row0 + r + half * 8) * f_out + col] = c[r];
}

// ---------------------------------------------------------------------------
// log_softmax over 64 cols: one wave per row, 2 values per lane
// ---------------------------------------------------------------------------
__global__ void logsoftmax_kernel(const float* __restrict__ in, float* __restrict__ out) {
  int gid = blockIdx.x * blockDim.x + threadIdx.x;
  int row = gid >> 5;
  int lane = gid & 31;
  if (row >= N_NODES) return;
  const float* r = in + (size_t)row * F_OUT_FINAL;
  float v0 = r[lane], v1 = r[lane + 32];
  float m = fmaxf(v0, v1);
#pragma unroll
  for (int off = 16; off > 0; off >>= 1) m = fmaxf(m, __shfl_xor(m, off, 32));
  float s = expf(v0 - m) + expf(v1 - m);
#pragma unroll
  for (int off = 16; off > 0; off >>= 1) s += __shfl_xor(s, off, 32);
  float ls = logf(s);
  float* o = out + (size_t)row * F_OUT_FINAL;
  o[lane]      = v0 - m - ls;
  o[lane + 32] = v1 - m - ls;
}

// ---------------------------------------------------------------------------
// Host-side orchestration
// ---------------------------------------------------------------------------
static inline int ceil_div(long long a, long long b) { return (int)((a + b - 1) / b); }

static void run_layer(const float* in, float* outb, float* t1, float* t2,
                      const float* Ws, const float* bias, int f_out,
                      const int* src, const int* dst, const float* norm,
                      hipStream_t stream) {
  const int n4 = N_NODES * F_FEAT / 4;
  const int ew_blocks = ceil_div(n4, 256);
  const int prop_blocks = ceil_div((long long)N_EDGES * 32, 256);
  const int tiles = (N_NODES / 16) * (f_out / 16);
  const int gemm_blocks = ceil_div(tiles, 8);         // 8 waves / block
  const size_t wstride = (size_t)F_FEAT * f_out;      // per-k weight matrix

  // term 0: out = T0 @ W0            (T0 = in)
  cheb_gemm_wmma<<<gemm_blocks, 256, 0, stream>>>(in, Ws + 0 * wstride, bias, outb,
                                                  F_FEAT, f_out, 0, 0);
  // T1 = prop(in)
  fill_zero_kernel<<<ew_blocks, 256, 0, stream>>>((float4*)t1, n4);
  prop_kernel<<<prop_blocks, 256, 0, stream>>>(src, dst, norm, in, t1, 1.0f);
  cheb_gemm_wmma<<<gemm_blocks, 256, 0, stream>>>(t1, Ws + 1 * wstride, bias, outb,
                                                  F_FEAT, f_out, 1, 0);
  // T2 = 2*prop(T1) - T0
  neg_copy_kernel<<<ew_blocks, 256, 0, stream>>>((float4*)t2, (const float4*)in, n4);
  prop_kernel<<<prop_blocks, 256, 0, stream>>>(src, dst, norm, t1, t2, 2.0f);
  cheb_gemm_wmma<<<gemm_blocks, 256, 0, stream>>>(t2, Ws + 2 * wstride, bias, outb,
                                                  F_FEAT, f_out, 1, 0);
  // T3 = 2*prop(T2) - T1   (negate T1 in place, scatter-add into it)
  neg_inplace_kernel<<<ew_blocks, 256, 0, stream>>>((float4*)t1, n4);
  prop_kernel<<<prop_blocks, 256, 0, stream>>>(src, dst, norm, t2, t1, 2.0f);
  cheb_gemm_wmma<<<gemm_blocks, 256, 0, stream>>>(t1, Ws + 3 * wstride, bias, outb,
                                                  F_FEAT, f_out, 1, 1); // + bias, ReLU
}

extern "C" void kernel_launch(void* const* d_in, const int* in_sizes, int n_in,
                              void* d_out, int out_size, void* d_ws, size_t ws_size,
                              hipStream_t stream) {
  const float* x   = (const float*)d_in[0];
  const int*   ei  = (const int*)d_in[1];
  const float* ea  = (const float*)d_in[2];
  const float* Ws1 = (const float*)d_in[3];
  const float* b1  = (const float*)d_in[4];
  const float* Ws2 = (const float*)d_in[5];
  const float* b2  = (const float*)d_in[6];
  const float* Ws3 = (const float*)d_in[7];
  const float* b3  = (const float*)d_in[8];
  const int* src = ei;              // edge_index[0]
  const int* dst = ei + N_EDGES;    // edge_index[1]

  float* ws   = (float*)d_ws;
  float* dis  = ws;  ws += N_NODES;                 // deg then rsqrt in place
  float* norm = ws;  ws += N_EDGES;
  float* bufA = ws;  ws += (size_t)N_NODES * F_FEAT;
  float* bufB = ws;  ws += (size_t)N_NODES * F_FEAT;   // T scratch 1
  float* bufC = ws;  ws += (size_t)N_NODES * F_FEAT;   // T scratch 2
  float* bufD = ws;  ws += (size_t)N_NODES * F_FEAT;

  // --- normalization ---
  fill_zero_kernel<<<ceil_div(N_NODES, 1024), 256, 0, stream>>>((float4*)dis, N_NODES / 4);
  deg_kernel<<<ceil_div(N_EDGES, 256), 256, 0, stream>>>(src, dst, ea, dis);
  dis_kernel<<<ceil_div(N_NODES, 256), 256, 0, stream>>>(dis);
  norm_kernel<<<ceil_div(N_EDGES, 256), 256, 0, stream>>>(src, dst, ea, dis, norm);

  // --- three ChebConv + ReLU layers ---
  run_layer(x,    bufD, bufB, bufC, Ws1, b1, 128, src, dst, norm, stream);
  run_layer(bufD, bufA, bufB, bufC, Ws2, b2, 128, src, dst, norm, stream);
  run_layer(bufA, bufD, bufB, bufC, Ws3, b3,  64, src, dst, norm, stream);

  // --- log_softmax over the 64 outputs ---
  logsoftmax_kernel<<<ceil_div((long long)N_NODES * 32, 256), 256, 0, stream>>>(
      bufD, (float*)d_out);
}